// OICR_83923660963975
// MI455X (gfx1250) — compile-verified
//
#include <hip/hip_runtime.h>
#include <hip/hip_bf16.h>

// Problem constants (fixed by the reference)
#define BS   8
#define PN   2048
#define CK   512
#define EPSC 1e-7f
#define IOUT 0.5f

typedef __attribute__((ext_vector_type(2))) float v2f;
typedef __attribute__((ext_vector_type(8))) float v8f;

// ---------------------------------------------------------------------------
// init: zero the atomic accumulators (ws + total). ws/d_out are poisoned 0xAA.
// ---------------------------------------------------------------------------
__global__ __launch_bounds__(32) void k_init(float* __restrict__ imb,
                                             float* __restrict__ total) {
  if (threadIdx.x < 4) imb[threadIdx.x] = 0.f;
  if (threadIdx.x == 0) *total = 0.f;
}

// ---------------------------------------------------------------------------
// Kernel 1: logit = softmax(inputs @ W + b, axis=-1)
// M=16384, K=512, N=4 — f32 WMMA 16x16x4, one 16-row tile per wave.
//   A 16x4 (M,K): lanes 0-15 hold K=0..1, lanes 16-31 hold K=2..3 (2 VGPRs)
//   B 4x16 (K,N): lanes 0-15 rows K=0/1, lanes 16-31 rows K=2/3; N>=4 zero
//   D 16x16: VGPR r -> row r + 8*(lane>=16), col = lane&15
// B fragments come from LDS (W transposed + a zero row) so the inner loop is
// branch-free: global_load_b64 (A) + ds_load_b64 (B) + v_wmma.
// ---------------------------------------------------------------------------
__global__ __launch_bounds__(256) void k1_gemm_softmax(
    const float* __restrict__ inputs,   // (BS*PN, CK)
    const float* __restrict__ W,        // (CK, 4)
    const float* __restrict__ bias,     // (4,)
    float* __restrict__ logit) {        // (BS*PN, 4)
  __shared__ float WT[5 * CK];              // WT[c][k]; row 4 = zeros (10 KB)
  __shared__ float tile_lds[8][16 * 16];    // per-wave D spill (8 KB)

  // Cooperative transpose of W into LDS, plus the zero row for lanes l>=4.
  for (int i = threadIdx.x; i < 4 * CK; i += 256) {
    const int c = i >> 9;                   // i / CK
    const int k = i & (CK - 1);             // i % CK
    WT[c * CK + k] = W[k * 4 + c];
  }
  for (int i = threadIdx.x; i < CK; i += 256) WT[4 * CK + i] = 0.f;
  __syncthreads();

  const int lane = threadIdx.x & 31;
  const int wave = threadIdx.x >> 5;
  const int tile = blockIdx.x * 8 + wave;   // 16-row tile
  const int row0 = tile * 16;
  const int half = lane >> 4;               // 0: K=0..1, 1: K=2..3
  const int l    = lane & 15;

  const float* arow = inputs + (size_t)(row0 + l) * CK + half * 2;
  const float* brow = WT + (l < 4 ? l : 4) * CK + half * 2;  // zero row if l>=4

  v8f acc = {};
  for (int k = 0; k < CK; k += 4) {
    v2f a  = *(const v2f*)(arow + k);       // global_load_b64
    v2f bm = *(const v2f*)(brow + k);       // ds_load_b64
    acc = __builtin_amdgcn_wmma_f32_16x16x4_f32(
        /*neg_a=*/false, a, /*neg_b=*/false, bm,
        /*c_mod=*/(short)0, acc, /*reuse_a=*/false, /*reuse_b=*/false);
  }

  // Spill D tile to LDS so one lane can see a whole row (4 valid columns).
  float* t = tile_lds[wave];
#pragma unroll
  for (int r = 0; r < 8; ++r) t[(r + half * 8) * 16 + l] = acc[r];
  __syncthreads();

  if (lane < 16) {
    const int m = lane;
    float x0 = t[m * 16 + 0] + bias[0];
    float x1 = t[m * 16 + 1] + bias[1];
    float x2 = t[m * 16 + 2] + bias[2];
    float x3 = t[m * 16 + 3] + bias[3];
    float mx = fmaxf(fmaxf(x0, x1), fmaxf(x2, x3));
    float e0 = expf(x0 - mx), e1 = expf(x1 - mx);
    float e2 = expf(x2 - mx), e3 = expf(x3 - mx);
    float inv = 1.f / (e0 + e1 + e2 + e3);
    float* o = logit + (size_t)(row0 + m) * 4;
    o[0] = e0 * inv; o[1] = e1 * inv; o[2] = e2 * inv; o[3] = e3 * inv;
  }
}

// ---------------------------------------------------------------------------
// Kernel 2a: am[b*3+c] = argmax_q pre_score[b,q,c] (first occurrence, like jnp)
// ---------------------------------------------------------------------------
__global__ __launch_bounds__(256) void k2a_argmax(
    const float* __restrict__ pre_score, int* __restrict__ am) {
  __shared__ float sval[256];
  __shared__ int   sidx[256];
  const int tid = threadIdx.x;
  const int b = blockIdx.x / 3;
  const int c = blockIdx.x % 3;

  float best = -1e30f;
  int bi = 0x7fffffff;
  for (int q = tid; q < PN; q += 256) {
    float v = pre_score[(size_t)(b * PN + q) * 4 + c];
    if (v > best || (v == best && q < bi)) { best = v; bi = q; }
  }
  sval[tid] = best; sidx[tid] = bi;
  __syncthreads();
  for (int s = 128; s > 0; s >>= 1) {
    if (tid < s) {
      float v2 = sval[tid + s]; int i2 = sidx[tid + s];
      if (v2 > sval[tid] || (v2 == sval[tid] && i2 < sidx[tid])) {
        sval[tid] = v2; sidx[tid] = i2;
      }
    }
    __syncthreads();
  }
  if (tid == 0) am[blockIdx.x] = sidx[0];
}

// ---------------------------------------------------------------------------
// Kernel 2: per (b,p): best-seed weight w, pseudo labels y_k, imbalance sums.
// Seeds(c,q) = label[c] && (ps3[b,c,q] > 0.5 || q == argmax). Candidates staged
// through LDS in 256-wide tiles. Tie-break = smallest flat index c*PN+q.
// ---------------------------------------------------------------------------
__device__ __forceinline__ float box_iou(float4 a, float4 b) {
  float area1 = (a.z - a.x) * (a.w - a.y);
  float area2 = (b.z - b.x) * (b.w - b.y);
  float lx = fmaxf(a.x, b.x), ly = fmaxf(a.y, b.y);
  float rx = fminf(a.z, b.z), ry = fminf(a.w, b.w);
  float iw = fmaxf(rx - lx, 0.f), ih = fmaxf(ry - ly, 0.f);
  float inter = iw * ih;
  return inter / (area1 + area2 - inter);
}

__global__ __launch_bounds__(256) void k2_seed_match(
    const float* __restrict__ pre_score,  // (BS,PN,4)
    const float* __restrict__ labels,     // (BS,4)
    const float* __restrict__ rois,       // (BS,PN,4)
    const int*   __restrict__ am,         // (BS,3)
    float* __restrict__ wgt,              // (BS*PN)
    float* __restrict__ yk,               // (BS*PN,4)
    float* __restrict__ imb) {            // (4,) atomic
  __shared__ float4 sroi[256];
  __shared__ float4 sps[256];
  __shared__ int    smask[256];
  __shared__ float  red[256];

  const int tid = threadIdx.x;
  const int b   = blockIdx.y;
  const int p   = blockIdx.x * 256 + tid;
  const float4* rois4 = (const float4*)rois;
  const float4* ps4   = (const float4*)pre_score;

  const float4 myroi = rois4[b * PN + p];
  const float lab0 = labels[b * 4 + 0];
  const float lab1 = labels[b * 4 + 1];
  const float lab2 = labels[b * 4 + 2];
  const int am0 = am[b * 3 + 0], am1 = am[b * 3 + 1], am2 = am[b * 3 + 2];

  float bestv = -1.f, bestw = 1.f;
  int bestflat = 0x7fffffff;
  int cover = 0;

  for (int qt = 0; qt < PN; qt += 256) {
    __syncthreads();                       // protect LDS from previous tile
    const int q = qt + tid;
    const float4 r4 = rois4[b * PN + q];
    const float4 p4 = ps4[b * PN + q];
    sroi[tid] = r4;
    sps[tid]  = p4;
    int m = 0;
    if (lab0 > 0.5f && (p4.x > 0.5f || q == am0)) m |= 1;
    if (lab1 > 0.5f && (p4.y > 0.5f || q == am1)) m |= 2;
    if (lab2 > 0.5f && (p4.z > 0.5f || q == am2)) m |= 4;
    smask[tid] = m;
    __syncthreads();

    if (qt + 256 < PN)                     // gfx1250: global_prefetch_b8
      __builtin_prefetch(&rois4[b * PN + qt + 256 + tid], 0, 1);

    for (int j = 0; j < 256; ++j) {
      const int mj = smask[j];
      if (!mj) continue;                   // not a seed for any active class
      const float iou = box_iou(myroi, sroi[j]);
      if (iou > IOUT) cover |= mj;
      const float4 pj = sps[j];
      const int qq = qt + j;
#pragma unroll
      for (int c = 0; c < 3; ++c) {
        if (mj & (1 << c)) {
          const int flat = c * PN + qq;
          const float psv = (c == 0) ? pj.x : (c == 1) ? pj.y : pj.z;
          if (iou > bestv || (iou == bestv && flat < bestflat)) {
            bestv = iou; bestflat = flat; bestw = psv;
          }
        }
      }
    }
  }

  const float w  = (bestflat != 0x7fffffff) ? bestw : 1.f;  // no seed -> 1
  const float y0 = (cover & 1) ? 1.f : 0.f;
  const float y1 = (cover & 2) ? 1.f : 0.f;
  const float y2 = (cover & 4) ? 1.f : 0.f;
  const float y3 = cover ? 0.f : 1.f;
  const int idx = b * PN + p;
  wgt[idx] = w;
  yk[idx * 4 + 0] = y0; yk[idx * 4 + 1] = y1;
  yk[idx * 4 + 2] = y2; yk[idx * 4 + 3] = y3;

  // imbalance[n] += sum_block y_n  (block reduce, then one atomic per block)
  const float yv[4] = {y0, y1, y2, y3};
#pragma unroll
  for (int n = 0; n < 4; ++n) {
    __syncthreads();
    red[tid] = yv[n];
    __syncthreads();
    for (int s = 128; s > 0; s >>= 1) {
      if (tid < s) red[tid] += red[tid + s];
      __syncthreads();
    }
    if (tid == 0) atomicAdd(&imb[n], red[0]);
  }
}

// ---------------------------------------------------------------------------
// Kernel 3: weighted focal loss, reduced to a scalar.
// ---------------------------------------------------------------------------
__global__ __launch_bounds__(256) void k3_loss(
    const float* __restrict__ logit, const float* __restrict__ wgt,
    const float* __restrict__ yk, const float* __restrict__ imb,
    const float* __restrict__ labels, float* __restrict__ total) {
  __shared__ float red[256];
  const int tid = threadIdx.x;
  const int idx = blockIdx.x * 256 + tid;
  const int b = idx / PN;
  const float wv = wgt[idx];

  float contrib = 0.f;
#pragma unroll
  for (int n = 0; n < 4; ++n) {
    const float lg = logit[idx * 4 + n];
    const float y  = yk[idx * 4 + n];
    const float lgc = fminf(fmaxf(lg, EPSC), 1.f - EPSC);
    const float om = 1.f - lgc;
    float loss = -y * logf(lgc) * om * om;          // gamma = 2
    loss = wv * loss / (imb[n] + 1e-7f);
    const float lab = (n == 3) ? 1.f : labels[b * 4 + n];
    const float wq = 10.f * expf(lg) * (1.f - lab) + lab;
    contrib += loss * wq;
  }

  red[tid] = contrib;
  __syncthreads();
  for (int s = 128; s > 0; s >>= 1) {
    if (tid < s) red[tid] += red[tid + s];
    __syncthreads();
  }
  if (tid == 0) atomicAdd(total, red[0] * (1.f / BS));
}

// ---------------------------------------------------------------------------
// Launch. d_in order: inputs, pre_score, labels, rois, W, b, num
// d_out: [logit (BS*PN*4) | total (1)]
// ---------------------------------------------------------------------------
extern "C" void kernel_launch(void* const* d_in, const int* in_sizes, int n_in,
                              void* d_out, int out_size, void* d_ws, size_t ws_size,
                              hipStream_t stream) {
  (void)in_sizes; (void)n_in; (void)out_size; (void)ws_size;
  const float* inputs    = (const float*)d_in[0];
  const float* pre_score = (const float*)d_in[1];
  const float* labels    = (const float*)d_in[2];
  const float* rois      = (const float*)d_in[3];
  const float* W         = (const float*)d_in[4];
  const float* bias      = (const float*)d_in[5];

  float* out   = (float*)d_out;
  float* logit = out;                 // BS*PN*4 = 65536 floats
  float* total = out + BS * PN * 4;   // 1 float

  // Workspace layout (floats): [0:4) imbalance | [4:28) argmax ints |
  // [32:32+BS*PN) w | then y_k (BS*PN*4)
  float* ws  = (float*)d_ws;
  float* imb = ws;
  int*   am  = (int*)(ws + 4);
  float* wgt = ws + 32;
  float* yk  = ws + 32 + BS * PN;

  k_init<<<1, 32, 0, stream>>>(imb, total);
  k1_gemm_softmax<<<dim3((BS * PN) / (16 * 8)), 256, 0, stream>>>(inputs, W, bias, logit);
  k2a_argmax<<<dim3(BS * 3), 256, 0, stream>>>(pre_score, am);
  k2_seed_match<<<dim3(PN / 256, BS), 256, 0, stream>>>(pre_score, labels, rois, am,
                                                        wgt, yk, imb);
  k3_loss<<<dim3((BS * PN) / 256), 256, 0, stream>>>(logit, wgt, yk, imb, labels, total);
}